// GCNLayer_21672404975784
// MI455X (gfx1250) — compile-verified
//
#include <hip/hip_runtime.h>
#include <hip/hip_bf16.h>
#include <stddef.h>

// ---------------------------------------------------------------------------
// GCN copy_u + segment_sum on MI455X (gfx1250).
//
// Roofline: 51.2 MFLOP vs ~450 MB moved (AI~0.2); features+out+edges fit in
// the 192MB L2, so this is L2-bandwidth/latency bound. WMMA is provably
// unhelpful (a one-hot SpMM wastes 16x the MACs and adds VALU work), so the
// optimization targets are: (1) kill the 51.2M fp32 atomics via a dst-CSR
// build, (2) wave32 lane==feature mapping -> every gather is one coalesced
// 128B line, (3) scalarize the wave-uniform index stream onto the SMEM path
// (s_load/KMcnt) so the vector pipe carries only feature data.
// ---------------------------------------------------------------------------

#define D_FEAT 32
#define SCAN_BLK 1024   // elements scanned per block in phase 3a

// ---- Phase 0: zero an int array (used for per-node counts) ----------------
__global__ void k_zero_i32(int* p, int n) {
    int i = blockIdx.x * blockDim.x + threadIdx.x;
    if (i < n) p[i] = 0;
}

// ---- Phase 1: histogram of destination nodes ------------------------------
__global__ void k_count(const int* __restrict__ dst, int* __restrict__ cnt, int E) {
    int e = blockIdx.x * blockDim.x + threadIdx.x;
    if (e < E) atomicAdd(&cnt[dst[e]], 1);
}

// ---- Phase 3a: per-block exclusive scan (in place: cnt -> off), block sums -
__global__ void k_scan_blocks(int* __restrict__ off, int* __restrict__ bsum, int n) {
    __shared__ int tmp[SCAN_BLK];
    int tid = threadIdx.x;
    int gid = blockIdx.x * SCAN_BLK + tid;
    int v = (gid < n) ? off[gid] : 0;
    tmp[tid] = v;
    __syncthreads();
    // Hillis-Steele inclusive scan in LDS
    for (int d = 1; d < SCAN_BLK; d <<= 1) {
        int t = (tid >= d) ? tmp[tid - d] : 0;
        __syncthreads();
        tmp[tid] += t;
        __syncthreads();
    }
    if (gid < n) off[gid] = tmp[tid] - v;                 // exclusive
    if (tid == SCAN_BLK - 1) bsum[blockIdx.x] = tmp[tid]; // block total
}

// ---- Phase 3b: exclusive scan of block sums (single block, nb <= 128) -----
__global__ void k_scan_bsum(int* __restrict__ bsum, int nb) {
    __shared__ int tmp[128];
    int tid = threadIdx.x;
    int v = (tid < nb) ? bsum[tid] : 0;
    tmp[tid] = v;
    __syncthreads();
    for (int d = 1; d < 128; d <<= 1) {
        int t = (tid >= d) ? tmp[tid - d] : 0;
        __syncthreads();
        tmp[tid] += t;
        __syncthreads();
    }
    if (tid < nb) bsum[tid] = tmp[tid] - v;               // exclusive
}

// ---- Phase 3c: add block offsets; init write cursors; seal off[n] ---------
__global__ void k_add_off(int* __restrict__ off, const int* __restrict__ bsum,
                          int* __restrict__ cur, int n, int E) {
    int gid = blockIdx.x * blockDim.x + threadIdx.x;
    if (gid < n) {
        int o = off[gid] + bsum[gid / SCAN_BLK];
        off[gid] = o;
        cur[gid] = o;
    }
    if (gid == 0) off[n] = E;
}

// ---- Phase 4: bucket edges by dst; store the SOURCE node id directly ------
__global__ void k_fill(const int* __restrict__ src, const int* __restrict__ dst,
                       int* __restrict__ cur, int* __restrict__ srclist, int E) {
    int e = blockIdx.x * blockDim.x + threadIdx.x;
    if (e < E) {
        int p = atomicAdd(&cur[dst[e]], 1);
        srclist[p] = src[e];
    }
}

// ---- Phase 5: one wave32 per node, lane == feature column -----------------
// node is wave-uniform (blockDim % 32 == 0); readfirstlane moves it (and
// therefore off[] and every srclist[] access) onto the scalar s_load path,
// leaving the vector memory pipe with only the coalesced 128B feature lines.
__global__ void k_gather_sum(const float* __restrict__ feat,
                             const int* __restrict__ off,
                             const int* __restrict__ srclist,
                             float* __restrict__ out, int n) {
    int gtid = blockIdx.x * blockDim.x + threadIdx.x;
    int node = __builtin_amdgcn_readfirstlane(gtid >> 5);  // wave-uniform -> SGPR
    int lane = threadIdx.x & 31;                           // feature column
    if (node >= n) return;
    int jb = off[node];        // s_load (uniform address)
    int je = off[node + 1];    // s_load
    float a0 = 0.f, a1 = 0.f, a2 = 0.f, a3 = 0.f;
    int j = jb;
    for (; j + 3 < je; j += 4) {
        // scalar index loads (s_load_b32, KMcnt) -- uniform address stream
        int s0 = srclist[j];
        int s1 = srclist[j + 1];
        int s2 = srclist[j + 2];
        int s3 = srclist[j + 3];
        // gfx1250 prefetch: pull the next batch's first row toward the WGP
        if (j + 7 < je)
            __builtin_prefetch(&feat[(size_t)srclist[j + 4] * D_FEAT], 0, 0);
        // four independent coalesced 128B gathers in flight
        a0 += feat[(size_t)s0 * D_FEAT + lane];
        a1 += feat[(size_t)s1 * D_FEAT + lane];
        a2 += feat[(size_t)s2 * D_FEAT + lane];
        a3 += feat[(size_t)s3 * D_FEAT + lane];
    }
    for (; j < je; ++j)
        a0 += feat[(size_t)srclist[j] * D_FEAT + lane];
    out[(size_t)node * D_FEAT + lane] = (a0 + a1) + (a2 + a3);
}

// ---- Fallback (tiny workspace): zero + direct fp32 atomic scatter ---------
__global__ void k_zero_f32(float* p, int n) {
    int i = blockIdx.x * blockDim.x + threadIdx.x;
    if (i < n) p[i] = 0.f;
}
__global__ void k_atomic_scatter(const float* __restrict__ feat,
                                 const int* __restrict__ src,
                                 const int* __restrict__ dst,
                                 float* __restrict__ out, int E) {
    int t = blockIdx.x * blockDim.x + threadIdx.x;
    int e = t >> 5;
    int f = t & 31;
    if (e < E) {
        float v = feat[(size_t)src[e] * D_FEAT + f];
        atomicAdd(&out[(size_t)dst[e] * D_FEAT + f], v);
    }
}

extern "C" void kernel_launch(void* const* d_in, const int* in_sizes, int n_in,
                              void* d_out, int out_size, void* d_ws, size_t ws_size,
                              hipStream_t stream) {
    const float* feat = (const float*)d_in[0];
    const int*   src  = (const int*)d_in[1];
    const int*   dst  = (const int*)d_in[2];
    float*       out  = (float*)d_out;

    const int N = in_sizes[0] / D_FEAT;   // 100000
    const int E = in_sizes[1];            // 1600000

    const int nScanBlocks = (N + SCAN_BLK - 1) / SCAN_BLK;   // 98 for N=100k

    // Workspace layout (int32): off[N+1] | cur[N] | bsum[128] | srclist[E]
    size_t need = ((size_t)(N + 1) + (size_t)N + 128 + (size_t)E) * sizeof(int);

    if (ws_size >= need && nScanBlocks <= 128) {
        int* off     = (int*)d_ws;
        int* cur     = off + (N + 1);
        int* bsum    = cur + N;
        int* srclist = bsum + 128;

        // Phase 0: zero counts (scan array reused for counts)
        k_zero_i32<<<(N + 1 + 255) / 256, 256, 0, stream>>>(off, N + 1);
        // Phase 1: per-node in-degree histogram (int atomics, L2-resident)
        k_count<<<(E + 255) / 256, 256, 0, stream>>>(dst, off, E);
        // Phase 3: exclusive prefix sum -> CSR row offsets
        k_scan_blocks<<<nScanBlocks, SCAN_BLK, 0, stream>>>(off, bsum, N);
        k_scan_bsum<<<1, 128, 0, stream>>>(bsum, nScanBlocks);
        k_add_off<<<(N + 255) / 256, 256, 0, stream>>>(off, bsum, cur, N, E);
        // Phase 4: bucket source ids by destination
        k_fill<<<(E + 255) / 256, 256, 0, stream>>>(src, dst, cur, srclist, E);
        // Phase 5: wave-per-node gather-sum, zero fp atomics, coalesced I/O
        {
            size_t threads = (size_t)N * 32;
            int grid = (int)((threads + 255) / 256);
            k_gather_sum<<<grid, 256, 0, stream>>>(feat, off, srclist, out, N);
        }
    } else {
        // Fallback: zero output then fp32 atomic scatter-add.
        k_zero_f32<<<(N * D_FEAT + 255) / 256, 256, 0, stream>>>(out, N * D_FEAT);
        size_t threads = (size_t)E * 32;
        int grid = (int)((threads + 255) / 256);
        k_atomic_scatter<<<grid, 256, 0, stream>>>(feat, src, dst, out, E);
    }
}